// SKRMLSTMCell_54288386622134
// MI455X (gfx1250) — compile-verified
//
#include <hip/hip_runtime.h>
#include <math.h>

// LSTM cell, D=8192: four GEMVs (1x16384)@(16384x8192) + gating.
// Memory-bound: 2.15 GB fp32 weights @ 23.3 TB/s ~= 92 us floor.
// Strategy: V_WMMA_F32_16X16X4_F32 accumulation, each weight byte read once.

#define DDIM   8192
#define K2     (2 * DDIM)          // 16384
#define TILE_N 16
#define NWAVES 16
#define BLOCK  (NWAVES * 32)       // 512 threads
#define KSLICES 4
#define KSLICE_LEN (K2 / KSLICES)  // 4096

typedef __attribute__((ext_vector_type(2))) float v2f;
typedef __attribute__((ext_vector_type(8))) float v8f;

__global__ __launch_bounds__(BLOCK)
void lstm_gemv_wmma(const float* __restrict__ x,  const float* __restrict__ h,
                    const float* __restrict__ c,
                    const float* __restrict__ Wc, const float* __restrict__ bc,
                    const float* __restrict__ Wf, const float* __restrict__ bf,
                    const float* __restrict__ Wi, const float* __restrict__ bi,
                    const float* __restrict__ Wo, const float* __restrict__ bo,
                    float* __restrict__ out)
{
    // 64KB LDS: concat = [h, x] during the GEMV loop; first 256 floats reused
    // afterwards (post-barrier) as the cross-wave reduction scratch.
    __shared__ __align__(16) float smem[K2];

    const int tid = threadIdx.x;

    // Stage concat into LDS with float4 loads (8 iterations per thread).
    {
        float4* s4 = reinterpret_cast<float4*>(smem);
        const float4* h4 = reinterpret_cast<const float4*>(h);
        const float4* x4 = reinterpret_cast<const float4*>(x);
        for (int i = tid; i < K2 / 4; i += BLOCK)
            s4[i] = (i < DDIM / 4) ? h4[i] : x4[i - DDIM / 4];
    }
    __syncthreads();

    const int lane = tid & 31;
    const int wid  = tid >> 5;      // 0..15
    const int gate = wid >> 2;      // 0..3 : c, f, i, o
    const int ksl  = wid & 3;       // 0..3 : K-slice of 4096
    const int j0   = blockIdx.x * TILE_N;
    const int lcol = lane & 15;
    const int lrow = (lane < 16) ? 0 : 2;   // B 4x16 layout: lanes<16 rows {k,k+1}, lanes>=16 rows {k+2,k+3}

    const float* Wg = (gate == 0) ? Wc : (gate == 1) ? Wf : (gate == 2) ? Wi : Wo;
    // Per-lane base pointer into this gate's weights for this block's column tile.
    const float* wptr = Wg + (size_t)(ksl * KSLICE_LEN + lrow) * DDIM + (size_t)(j0 + lcol);
    const float* ccp  = smem + ksl * KSLICE_LEN;

    v8f acc = {};   // D/C 16x16 f32; only row M=0 (VGPR0, lanes 0-15) is meaningful.

#pragma unroll 4
    for (int k = 0; k < KSLICE_LEN; k += 4) {
        // A-matrix 16x4: row 0 = concat[k0..k0+3], rows 1..15 = 0.
        // ISA layout: v0 lane0=K0, lane16=K2 ; v1 lane0=K1, lane16=K3.
        float4 cv = *reinterpret_cast<const float4*>(ccp + k);  // LDS broadcast
        v2f a, b;
        a.x = (lane == 0) ? cv.x : (lane == 16) ? cv.z : 0.0f;
        a.y = (lane == 0) ? cv.y : (lane == 16) ? cv.w : 0.0f;

        // B-matrix 4x16 weight tile, rows k..k+3, cols j0..j0+15 (coalesced 64B runs).
        const float* wk = wptr + (size_t)k * DDIM;
        b.x = __builtin_nontemporal_load(wk);          // row k0 + lrow
        b.y = __builtin_nontemporal_load(wk + DDIM);   // row k0 + lrow + 1

        // D = A*B + C, chained accumulation over K.
        acc = __builtin_amdgcn_wmma_f32_16x16x4_f32(
                  false, a, false, b, (short)0, acc, false, false);
    }

    __syncthreads();   // all waves done reading the concat region of smem

    // Each wave deposits its 16 partial sums (D row 0 = acc[0] on lanes 0-15).
    if (lane < 16)
        smem[wid * TILE_N + lane] = acc[0];
    __syncthreads();

    // 16 threads finish: sum 4 K-slices per gate, add bias, LSTM combine.
    if (tid < TILE_N) {
        const int j = j0 + tid;
        float gc = smem[(0 * 4 + 0) * 16 + tid] + smem[(0 * 4 + 1) * 16 + tid]
                 + smem[(0 * 4 + 2) * 16 + tid] + smem[(0 * 4 + 3) * 16 + tid] + bc[j];
        float gf = smem[(1 * 4 + 0) * 16 + tid] + smem[(1 * 4 + 1) * 16 + tid]
                 + smem[(1 * 4 + 2) * 16 + tid] + smem[(1 * 4 + 3) * 16 + tid] + bf[j];
        float gi = smem[(2 * 4 + 0) * 16 + tid] + smem[(2 * 4 + 1) * 16 + tid]
                 + smem[(2 * 4 + 2) * 16 + tid] + smem[(2 * 4 + 3) * 16 + tid] + bi[j];
        float go = smem[(3 * 4 + 0) * 16 + tid] + smem[(3 * 4 + 1) * 16 + tid]
                 + smem[(3 * 4 + 2) * 16 + tid] + smem[(3 * 4 + 3) * 16 + tid] + bo[j];

        float z  = tanhf(gc);
        float zf = 1.0f / (1.0f + expf(-gf));
        float zi = 1.0f / (1.0f + expf(-gi));
        float zo = 1.0f / (1.0f + expf(-go));
        float cn = c[j] * zf + z * zi;
        out[j] = zo * tanhf(cn);
    }
}

extern "C" void kernel_launch(void* const* d_in, const int* in_sizes, int n_in,
                              void* d_out, int out_size, void* d_ws, size_t ws_size,
                              hipStream_t stream) {
    (void)in_sizes; (void)n_in; (void)d_ws; (void)ws_size; (void)out_size;
    const float* x  = (const float*)d_in[0];
    const float* h  = (const float*)d_in[1];
    const float* c  = (const float*)d_in[2];
    const float* Wc = (const float*)d_in[3];
    const float* bc = (const float*)d_in[4];
    const float* Wf = (const float*)d_in[5];
    const float* bf = (const float*)d_in[6];
    const float* Wi = (const float*)d_in[7];
    const float* bi = (const float*)d_in[8];
    const float* Wo = (const float*)d_in[9];
    const float* bo = (const float*)d_in[10];
    float* out = (float*)d_out;

    dim3 grid(DDIM / TILE_N);   // 512 column tiles
    dim3 block(BLOCK);          // 16 waves: 4 gates x 4 K-slices
    lstm_gemv_wmma<<<grid, block, 0, stream>>>(x, h, c, Wc, bc, Wf, bf, Wi, bi, Wo, bo, out);
}